// FeatureTransformerSlice_46660524703857
// MI455X (gfx1250) — compile-verified
//
#include <hip/hip_runtime.h>
#include <hip/hip_bf16.h>
#include <stdint.h>

// ---------------------------------------------------------------------------
// Sparse feature transformer: out[b,:] = bias + sum_k fval[b,k] * W[fidx[b,k],:]
// B=16384, K=32 active rows, O=512 fp32 outputs, W is 45056x512 (92MB -> fits L2).
//
// CDNA5 path: Tensor Data Mover in gather mode. One D# gathers 16 rows
// (16-bit indices) from the weight tensor into LDS asynchronously
// (TENSORcnt). Two TDM ops stage all 32 rows (64KB LDS) while other resident
// workgroups on the WGP keep the VALUs busy with the FMA reduction.
// ---------------------------------------------------------------------------

typedef __attribute__((ext_vector_type(4))) unsigned int u32x4;
typedef __attribute__((ext_vector_type(8))) int          i32x8;
typedef __attribute__((ext_vector_type(4))) int          i32x4;

#define FT_NUM_INPUTS  45056
#define FT_NUM_OUTPUTS 512
#define FT_MAX_ACTIVE  32
#define FT_BLOCK       256   // 8 wave32; each thread owns 2 consecutive outputs

#define FT_USE_TDM (__has_builtin(__builtin_amdgcn_tensor_load_to_lds))

__global__ __launch_bounds__(FT_BLOCK)
void ft_gather_tdm_kernel(const int*   __restrict__ fidx,
                          const float* __restrict__ fval,
                          const float* __restrict__ weight,
                          const float* __restrict__ bias,
                          float*       __restrict__ out)
{
    const int b    = blockIdx.x;
    const int t    = threadIdx.x;
    const int lane = t & 31;

    // Every wave holds the 32 per-feature scales, one per lane; broadcast with
    // v_readlane inside the k-loop (no redundant per-k scalar loads).
    const float vlane = fval[b * FT_MAX_ACTIVE + lane];

#if FT_USE_TDM
    __shared__ float lds_rows[FT_MAX_ACTIVE * FT_NUM_OUTPUTS]; // 64 KB

    if (t < 32) {
        // Wave 0: build TDM gather descriptors in SGPRs and fire two DMAs.
        const int my_idx = fidx[b * FT_MAX_ACTIVE + lane];

        const uint64_t waddr    = (uint64_t)(uintptr_t)weight;
        const uint32_t lds_base = (uint32_t)(uintptr_t)lds_rows; // low 32b of LDS-aperture flat addr = LDS offset

        // ---- D# group 1: tensor/tile geometry (shared by both ops) ----
        i32x8 g1;
        g1[0] = 0x00020000;                               // data_size=2 (4-byte elements); no multicast/pad/iterate
        g1[1] = (int)((unsigned)FT_NUM_OUTPUTS << 16);    // tensor_dim0 = 512 (row length), bits[79:48]
        g1[2] = (int)(((unsigned)FT_NUM_INPUTS & 0xFFFFu) << 16); // tensor_dim1 = 45056 rows, bits[111:80]
        g1[3] = (int)((unsigned)FT_NUM_OUTPUTS << 16);    // tile_dim0 = 512, bits[127:112]
        g1[4] = 16;                                       // tile_dim1 = #valid gather indices
        g1[5] = FT_NUM_OUTPUTS;                           // tensor_dim0_stride = 512 elements
        g1[6] = 0;
        g1[7] = 0;

        // Pack 32 row indices as 16-bit pairs (scalar, via v_readlane).
        int p[16];
#pragma unroll
        for (int i = 0; i < 16; ++i) {
            const int lo = __builtin_amdgcn_readlane(my_idx, 2 * i);
            const int hi = __builtin_amdgcn_readlane(my_idx, 2 * i + 1);
            p[i] = (lo & 0xFFFF) | (hi << 16);
        }

        // ---- D# group 0: control / addresses ----
        // [1:0] count=1, [30] gather_index_size=0 (16-bit), [31] gather_mode=1
        u32x4 g0;
        g0[0] = 0x80000001u;
        g0[1] = lds_base;                                   // rows 0..15 -> LDS offset 0
        g0[2] = (uint32_t)(waddr & 0xFFFFFFFFu);            // global_addr[31:0]
        g0[3] = (uint32_t)((waddr >> 32) & 0x1FFFFFFu)      // global_addr[56:32]
              | (2u << 30);                                 // type = 2 ("image")

        // amdgpu-toolchain (clang-23) builtin is the 6-arg form:
        //   (u32x4 g0, i32x8 g1, i32x4 g2, i32x4 g3, i32x8 extra, i32 cpol)
        // args 0..3 = D# SGPR groups 0..3 per the ISA; extra group zero-filled.
        const i32x8 gz = { 0, 0, 0, 0, 0, 0, 0, 0 };

        i32x4 g2a = { p[0], p[1], p[2], p[3] };
        i32x4 g3a = { p[4], p[5], p[6], p[7] };
        __builtin_amdgcn_tensor_load_to_lds(g0, g1, g2a, g3a, gz, 0);

        u32x4 g0b = g0;
        g0b[1] = lds_base + 16u * FT_NUM_OUTPUTS * sizeof(float); // rows 16..31 -> +32KB
        i32x4 g2b = { p[8],  p[9],  p[10], p[11] };
        i32x4 g3b = { p[12], p[13], p[14], p[15] };
        __builtin_amdgcn_tensor_load_to_lds(g0b, g1, g2b, g3b, gz, 0);

        __builtin_amdgcn_s_wait_tensorcnt(0);               // all gathered rows landed in LDS
    }
    __syncthreads();                                        // make LDS rows visible to all 8 waves

    // ---- Reduction: each thread owns outputs {2t, 2t+1} ----
    float2 acc = ((const float2*)bias)[t];
#pragma unroll
    for (int k = 0; k < FT_MAX_ACTIVE; ++k) {
        const float vk =
            __int_as_float(__builtin_amdgcn_readlane(__float_as_int(vlane), k));
        // 32 lanes x consecutive float2 => 64 consecutive dwords => all 64 LDS banks, conflict-free
        const float2 w = ((const float2*)(lds_rows + k * FT_NUM_OUTPUTS))[t];
        acc.x += vk * w.x;
        acc.y += vk * w.y;
    }
#else
    // Fallback: direct L2-served gather (weights fit in 192MB L2).
    __shared__ int sidx[FT_MAX_ACTIVE];
    if (t < 32) sidx[t] = fidx[b * FT_MAX_ACTIVE + t];
    __syncthreads();

    float2 acc = ((const float2*)bias)[t];
#pragma unroll
    for (int k = 0; k < FT_MAX_ACTIVE; ++k) {
        const float vk =
            __int_as_float(__builtin_amdgcn_readlane(__float_as_int(vlane), k));
        const float2 w =
            ((const float2*)(weight + (size_t)sidx[k] * FT_NUM_OUTPUTS))[t];
        acc.x += vk * w.x;
        acc.y += vk * w.y;
    }
#endif

    ((float2*)out)[(size_t)b * (FT_NUM_OUTPUTS / 2) + t] = acc;
}

extern "C" void kernel_launch(void* const* d_in, const int* in_sizes, int n_in,
                              void* d_out, int out_size, void* d_ws, size_t ws_size,
                              hipStream_t stream) {
    const int*   fidx   = (const int*)d_in[0];    // [B, 32] int32
    const float* fval   = (const float*)d_in[1];  // [B, 32] f32
    const float* weight = (const float*)d_in[2];  // [45056, 512] f32
    const float* bias   = (const float*)d_in[3];  // [512] f32
    float*       out    = (float*)d_out;          // [B, 512] f32

    const int batch = in_sizes[0] / FT_MAX_ACTIVE;

    ft_gather_tdm_kernel<<<batch, FT_BLOCK, 0, stream>>>(fidx, fval, weight, bias, out);
}